// BayesianTripletLoss_82746839925070
// MI455X (gfx1250) — compile-verified
//
#include <hip/hip_runtime.h>
#include <math.h>

typedef float v2f __attribute__((ext_vector_type(2)));
typedef float v8f __attribute__((ext_vector_type(8)));

#define B_SZ 512
#define D_SZ 256
#define MARGIN 0.3f
#define UW 0.05f
#define MIN_U 1e-6f
#define MAX_U 1.0f
#define EPS 1e-8f

// ---------------- kernel 1: row norms ‖e_i‖^2 ----------------
__global__ __launch_bounds__(256) void norms_kernel(const float* __restrict__ E,
                                                    float* __restrict__ norms) {
    __shared__ float red[256];
    int row = blockIdx.x;
    int t = threadIdx.x;
    float v = E[row * D_SZ + t];
    red[t] = v * v;
    __syncthreads();
    for (int s = 128; s > 0; s >>= 1) {
        if (t < s) red[t] += red[t + s];
        __syncthreads();
    }
    if (t == 0) norms[row] = red[0];
}

// ---------------- kernel 2: E @ E^T via V_WMMA_F32_16X16X4_F32, fused dist ----------------
// One wave per 16x16 output tile. A layout (32-bit, 16x4): lane<16 -> row=lane, K={0,1};
// lane>=16 -> row=lane-16, K={2,3}. B (4x16) mirrors with N=lane&15. Both sides read rows
// of E (symmetric GEMM), so each lane loads one float2 per matrix per k-step.
__global__ __launch_bounds__(256) void gram_kernel(const float* __restrict__ E,
                                                   const float* __restrict__ norms,
                                                   float* __restrict__ dist) {
    int wave = (blockIdx.x * blockDim.x + threadIdx.x) >> 5;  // 1024 tiles
    int lane = threadIdx.x & 31;
    int tm = wave >> 5;        // tile row  (32 tiles)
    int tn = wave & 31;        // tile col
    int ib = tm * 16, jb = tn * 16;
    int half = lane >> 4;
    int r16 = lane & 15;

    const float* arow = E + (ib + r16) * D_SZ + 2 * half;
    const float* brow = E + (jb + r16) * D_SZ + 2 * half;

    v8f acc = {};
#pragma unroll 8
    for (int k = 0; k < D_SZ; k += 4) {
        v2f a = *(const v2f*)(arow + k);
        v2f b = *(const v2f*)(brow + k);
        acc = __builtin_amdgcn_wmma_f32_16x16x4_f32(
            /*neg_a=*/false, a, /*neg_b=*/false, b,
            /*c_mod=*/(short)0, acc, /*reuse_a=*/false, /*reuse_b=*/false);
    }

    float nj = norms[jb + r16];  // column norm for this lane's N
#pragma unroll
    for (int r = 0; r < 8; ++r) {
        int m = r + 8 * half;  // C/D layout: VGPR r -> M=r (lanes 0-15) / M=r+8 (lanes 16-31)
        float ni = norms[ib + m];
        float d2 = ni + nj - 2.0f * acc[r];
        d2 = d2 > 0.0f ? d2 : 0.0f;
        dist[(ib + m) * B_SZ + (jb + r16)] = sqrtf(d2) + EPS;
    }
}

// ---------------- kernel 3: hardest positive / negative per anchor row ----------------
__global__ __launch_bounds__(256) void mine_kernel(const float* __restrict__ dist,
                                                   const int* __restrict__ labels,
                                                   int* __restrict__ hp, int* __restrict__ hn,
                                                   int* __restrict__ validA) {
    int row = (blockIdx.x * blockDim.x + threadIdx.x) >> 5;  // one wave per row
    int lane = threadIdx.x & 31;
    int myLabel = labels[row];

    float bestP = -1e30f; int bpi = 0;
    float bestN =  1e30f; int bni = 0;
    for (int j = lane; j < B_SZ; j += 32) {
        float d = dist[row * B_SZ + j];
        bool same = (labels[j] == myLabel);
        if (same && j != row) {
            if (d > bestP) { bestP = d; bpi = j; }
        }
        if (!same) {
            if (d < bestN) { bestN = d; bni = j; }
        }
    }
    for (int off = 16; off > 0; off >>= 1) {
        float oP = __shfl_down(bestP, off, 32); int oPi = __shfl_down(bpi, off, 32);
        float oN = __shfl_down(bestN, off, 32); int oNi = __shfl_down(bni, off, 32);
        if (oP > bestP) { bestP = oP; bpi = oPi; }
        if (oN < bestN) { bestN = oN; bni = oNi; }
    }
    if (lane == 0) {
        hp[row] = bpi;
        hn[row] = bni;
        validA[row] = (bestP > -1e29f && bestN < 1e29f) ? 1 : 0;
    }
}

// ---------------- kernel 4: selected-pair uncertainties + loss reduction ----------------
__global__ __launch_bounds__(512) void finalize_kernel(const float* __restrict__ E,
                                                       const float* __restrict__ U,
                                                       const float* __restrict__ dist,
                                                       const int* __restrict__ hp,
                                                       const int* __restrict__ hn,
                                                       const int* __restrict__ validA,
                                                       float* __restrict__ out) {
    __shared__ float red[512];
    int t = threadIdx.x;

    // sum of clipped uncertainties (regularizer)
    float usum = 0.0f;
    for (int idx = t; idx < B_SZ * D_SZ; idx += 512) {
        float u = U[idx];
        u = fminf(fmaxf(u, MIN_U), MAX_U);  // NaN -> MIN_U via fmaxf semantics
        usum += u;
    }

    // one anchor per thread: dist_unc only at the two mined pairs
    int i = t;
    int jp = hp[i], jn = hn[i];
    int vld = validA[i];
    const float* ei = E + i * D_SZ;
    const float* ui = U + i * D_SZ;
    const float* ep = E + jp * D_SZ;
    const float* en = E + jn * D_SZ;
    float Sp = 0.0f, Sn = 0.0f;
    for (int d = 0; d < D_SZ; ++d) {
        float u = ui[d];
        u = fminf(fmaxf(u, MIN_U), MAX_U);
        float w = u * u;
        float e = ei[d];
        float dp = e - ep[d];
        float dn = e - en[d];
        Sp += dp * dp * w;
        Sn += dn * dn * w;
    }
    float d_pos = dist[i * B_SZ + jp];
    float d_neg = dist[i * B_SZ + jn];
    float u_pos = sqrtf(Sp / (d_pos * d_pos) + EPS);
    float u_neg = sqrtf(Sn / (d_neg * d_neg) + EPS);
    float sigma = sqrtf(u_pos * u_pos + u_neg * u_neg + EPS);
    float margin_eff = MARGIN + UW * sigma;
    float z = (d_pos - d_neg + margin_eff) / sigma;  // TEMPERATURE = 1
    float sp = fmaxf(z, 0.0f) + log1pf(expf(-fabsf(z)));  // stable softplus
    float lossC = vld ? (sigma * sp) : 0.0f;
    float cnt = vld ? 1.0f : 0.0f;

    // reduce loss
    red[t] = lossC; __syncthreads();
    for (int s = 256; s > 0; s >>= 1) { if (t < s) red[t] += red[t + s]; __syncthreads(); }
    float sumLoss = red[0]; __syncthreads();
    // reduce valid count
    red[t] = cnt; __syncthreads();
    for (int s = 256; s > 0; s >>= 1) { if (t < s) red[t] += red[t + s]; __syncthreads(); }
    float nvalid = red[0]; __syncthreads();
    // reduce u sum
    red[t] = usum; __syncthreads();
    for (int s = 256; s > 0; s >>= 1) { if (t < s) red[t] += red[t + s]; __syncthreads(); }

    if (t == 0) {
        float mainL = sumLoss / fmaxf(nvalid, 1.0f);
        float total = mainL + UW * (red[0] / (float)(B_SZ * D_SZ));
        if (isnan(total) || isinf(total)) total = 0.0f;
        out[0] = total;
    }
}

extern "C" void kernel_launch(void* const* d_in, const int* in_sizes, int n_in,
                              void* d_out, int out_size, void* d_ws, size_t ws_size,
                              hipStream_t stream) {
    (void)in_sizes; (void)n_in; (void)out_size; (void)ws_size;
    const float* E = (const float*)d_in[0];
    const float* U = (const float*)d_in[1];
    const int* labels = (const int*)d_in[2];
    float* out = (float*)d_out;

    char* ws = (char*)d_ws;
    float* norms = (float*)ws;                              // 512 f32
    float* dist  = (float*)(ws + 4096);                     // 512*512 f32 (1 MB)
    int*   hp    = (int*)(ws + 4096 + B_SZ * B_SZ * 4);     // 512 i32
    int*   hn    = hp + B_SZ;
    int*   vld   = hn + B_SZ;

    norms_kernel<<<B_SZ, 256, 0, stream>>>(E, norms);
    gram_kernel<<<128, 256, 0, stream>>>(E, norms, dist);   // 1024 waves = 1024 tiles
    mine_kernel<<<64, 256, 0, stream>>>(dist, labels, hp, hn, vld);
    finalize_kernel<<<1, 512, 0, stream>>>(E, U, dist, hp, hn, vld, out);
}